// SAGELayer2_7241314861140
// MI455X (gfx1250) — compile-verified
//
#include <hip/hip_runtime.h>

typedef __attribute__((ext_vector_type(2))) float v2f;
typedef __attribute__((ext_vector_type(8))) float v8f;

#define LDSP 130  // padded LDS pitch (dwords) for 128-wide weight rows: stride 130 % 64 = 2 -> conflict-free

// ---------------------------------------------------------------- zero scratch
__global__ void zero_ws_kernel(float4* __restrict__ p, int n4) {
  int i = blockIdx.x * blockDim.x + threadIdx.x;
  int stride = gridDim.x * blockDim.x;
  float4 z = make_float4(0.f, 0.f, 0.f, 0.f);
  for (; i < n4; i += stride) p[i] = z;
}

// ---------------------------------------------------------------- edge scatter
// one 32-lane wave per edge; each lane handles 2 consecutive features
__global__ void scatter_kernel(const float* __restrict__ efeats,
                               const int* __restrict__ u,
                               const int* __restrict__ v,
                               float* __restrict__ sums_v,
                               float* __restrict__ sums_u,
                               float* __restrict__ deg_v,
                               float* __restrict__ deg_u,
                               int E) {
  int gid = blockIdx.x * blockDim.x + threadIdx.x;
  int e = gid >> 5;
  if (e >= E) return;
  int lane = gid & 31;
  int ve = v[e];
  int ue = u[e];
  const float* src = efeats + (size_t)e * 64 + lane * 2;
  float a0 = src[0], a1 = src[1];
  float* pv = sums_v + (size_t)ve * 64 + lane * 2;
  float* pu = sums_u + (size_t)ue * 64 + lane * 2;
  atomicAdd(pv + 0, a0);
  atomicAdd(pv + 1, a1);
  atomicAdd(pu + 0, a0);
  atomicAdd(pu + 1, a1);
  if (lane == 0) {
    atomicAdd(deg_v + ve, 1.0f);
    atomicAdd(deg_u + ue, 1.0f);
  }
}

// ---------------------------------------------------------------- WMMA helper
__device__ __forceinline__ v8f wmma_f32(v2f a, v2f b, v8f c) {
  // V_WMMA_F32_16X16X4_F32: D = A(16x4,f32) x B(4x16,f32) + C(16x16,f32)
  return __builtin_amdgcn_wmma_f32_16x16x4_f32(false, a, false, b, (short)0, c,
                                               false, false);
}

// ---------------------------------------------------------------- node GEMM
// h = relu([sums_v/deg_v | sums_u/deg_u] @ Wn^T + bn), one wave per 16 nodes
__global__ __launch_bounds__(64) void node_gemm_kernel(
    const float* __restrict__ sums_v, const float* __restrict__ sums_u,
    const float* __restrict__ deg_v, const float* __restrict__ deg_u,
    const float* __restrict__ Wn, const float* __restrict__ bn,
    float* __restrict__ h) {
  __shared__ float W[64 * LDSP];
  const int tid = threadIdx.x;  // 64 threads = 2 waves
  for (int i = tid; i < 2048; i += 64) {  // 64 rows x 32 float4
    int row = i >> 5, c4 = (i & 31) << 2;
    const float4 w = *(const float4*)(Wn + row * 128 + c4);
    float* d = &W[row * LDSP + c4];
    d[0] = w.x; d[1] = w.y; d[2] = w.z; d[3] = w.w;
  }
  __syncthreads();

  const int lane = tid & 31;
  const int tile = blockIdx.x * 2 + (tid >> 5);  // 16-node tile
  const int m = lane & 15;                       // A row / B col in-lane index
  const int koff = (lane >> 4) << 1;             // K sub-offset: 0 or 2
  const int node = tile * 16 + m;

  const float rcv = 1.0f / fmaxf(deg_v[node], 1.0f);
  const float rcu = 1.0f / fmaxf(deg_u[node], 1.0f);
  const float* __restrict__ sv = sums_v + (size_t)node * 64 + koff;
  const float* __restrict__ su = sums_u + (size_t)node * 64 + koff;

  v8f acc0 = {}, acc1 = {}, acc2 = {}, acc3 = {};
#pragma unroll
  for (int k = 0; k < 128; k += 4) {
    v2f a;
    if (k < 64) {
      a = *(const v2f*)(sv + k);
      a *= rcv;
    } else {
      a = *(const v2f*)(su + (k - 64));
      a *= rcu;
    }
    const int kb = k + koff;
    v2f b0 = *(const v2f*)(&W[(0 * 16 + m) * LDSP + kb]);
    v2f b1 = *(const v2f*)(&W[(1 * 16 + m) * LDSP + kb]);
    v2f b2 = *(const v2f*)(&W[(2 * 16 + m) * LDSP + kb]);
    v2f b3 = *(const v2f*)(&W[(3 * 16 + m) * LDSP + kb]);
    acc0 = wmma_f32(a, b0, acc0);
    acc1 = wmma_f32(a, b1, acc1);
    acc2 = wmma_f32(a, b2, acc2);
    acc3 = wmma_f32(a, b3, acc3);
  }

  const int rsel = (lane >> 4) << 3;  // row select: 0 or 8
  const float bn0 = bn[0 + m], bn1 = bn[16 + m], bn2 = bn[32 + m],
              bn3 = bn[48 + m];
#pragma unroll
  for (int i = 0; i < 8; ++i) {
    float* dst = h + (size_t)(tile * 16 + i + rsel) * 64;
    dst[ 0 + m] = fmaxf(acc0[i] + bn0, 0.f);
    dst[16 + m] = fmaxf(acc1[i] + bn1, 0.f);
    dst[32 + m] = fmaxf(acc2[i] + bn2, 0.f);
    dst[48 + m] = fmaxf(acc3[i] + bn3, 0.f);
  }
}

// ---------------------------------------------------------------- edge GEMM
// edge = [h[u[e]] | efeats[e]] @ We^T + be, one wave per 16 edges
__global__ __launch_bounds__(256) void edge_gemm_kernel(
    const float* __restrict__ h, const float* __restrict__ efeats,
    const int* __restrict__ u, const float* __restrict__ We,
    const float* __restrict__ be, float* __restrict__ out_edge) {
  __shared__ float W[64 * LDSP];
  const int tid = threadIdx.x;  // 256 threads = 8 waves
  for (int i = tid; i < 2048; i += 256) {
    int row = i >> 5, c4 = (i & 31) << 2;
    const float4 w = *(const float4*)(We + row * 128 + c4);
    float* d = &W[row * LDSP + c4];
    d[0] = w.x; d[1] = w.y; d[2] = w.z; d[3] = w.w;
  }
  __syncthreads();

  const int lane = tid & 31;
  const int tile = blockIdx.x * 8 + (tid >> 5);  // 16-edge tile
  const int m = lane & 15;
  const int koff = (lane >> 4) << 1;
  const int e = tile * 16 + m;
  const int ue = u[e];
  const float* __restrict__ ha = h + (size_t)ue * 64 + koff;       // gather
  const float* __restrict__ ea = efeats + (size_t)e * 64 + koff;

  v8f acc0 = {}, acc1 = {}, acc2 = {}, acc3 = {};
#pragma unroll
  for (int k = 0; k < 128; k += 4) {
    v2f a = (k < 64) ? *(const v2f*)(ha + k) : *(const v2f*)(ea + (k - 64));
    const int kb = k + koff;
    v2f b0 = *(const v2f*)(&W[(0 * 16 + m) * LDSP + kb]);
    v2f b1 = *(const v2f*)(&W[(1 * 16 + m) * LDSP + kb]);
    v2f b2 = *(const v2f*)(&W[(2 * 16 + m) * LDSP + kb]);
    v2f b3 = *(const v2f*)(&W[(3 * 16 + m) * LDSP + kb]);
    acc0 = wmma_f32(a, b0, acc0);
    acc1 = wmma_f32(a, b1, acc1);
    acc2 = wmma_f32(a, b2, acc2);
    acc3 = wmma_f32(a, b3, acc3);
  }

  const int rsel = (lane >> 4) << 3;
  const float be0 = be[0 + m], be1 = be[16 + m], be2 = be[32 + m],
              be3 = be[48 + m];
#pragma unroll
  for (int i = 0; i < 8; ++i) {
    float* dst = out_edge + (size_t)(tile * 16 + i + rsel) * 64;
    dst[ 0 + m] = acc0[i] + be0;
    dst[16 + m] = acc1[i] + be1;
    dst[32 + m] = acc2[i] + be2;
    dst[48 + m] = acc3[i] + be3;
  }
}

// ---------------------------------------------------------------- launcher
extern "C" void kernel_launch(void* const* d_in, const int* in_sizes, int n_in,
                              void* d_out, int out_size, void* d_ws,
                              size_t ws_size, hipStream_t stream) {
  // setup_inputs order: nfeats, efeats, u, v, Wn, bn, We, be  (nfeats unused)
  const float* efeats = (const float*)d_in[1];
  const int* u = (const int*)d_in[2];
  const int* v = (const int*)d_in[3];
  const float* Wn = (const float*)d_in[4];
  const float* bn = (const float*)d_in[5];
  const float* We = (const float*)d_in[6];
  const float* be = (const float*)d_in[7];

  const int N = in_sizes[0] / 64;  // 100000
  const int E = in_sizes[2];       // 1600000

  float* h_out = (float*)d_out;                       // [N,64]
  float* edge_out = (float*)d_out + (size_t)N * 64;   // [E,64]

  // workspace: sums_v[N*64] | sums_u[N*64] | deg_v[N] | deg_u[N]  (52 MB)
  float* sums_v = (float*)d_ws;
  float* sums_u = sums_v + (size_t)N * 64;
  float* deg_v = sums_u + (size_t)N * 64;
  float* deg_u = deg_v + N;

  const int total_f = N * 128 + 2 * N;  // divisible by 4
  zero_ws_kernel<<<1024, 256, 0, stream>>>((float4*)d_ws, total_f / 4);

  const long long sthreads = (long long)E * 32;
  scatter_kernel<<<(int)((sthreads + 255) / 256), 256, 0, stream>>>(
      efeats, u, v, sums_v, sums_u, deg_v, deg_u, E);

  // N/16 = 6250 tiles, 2 waves (of 32) per block of 64 -> 3125 blocks
  node_gemm_kernel<<<N / 32, 64, 0, stream>>>(sums_v, sums_u, deg_v, deg_u, Wn,
                                              bn, h_out);

  // E/16 = 100000 tiles, 8 waves per block of 256 -> 12500 blocks
  edge_gemm_kernel<<<E / 128, 256, 0, stream>>>(h_out, efeats, u, We, be,
                                                edge_out);
}